// GraphAttNet_31817117729462
// MI455X (gfx1250) — compile-verified
//
#include <hip/hip_runtime.h>
#include <hip/hip_bf16.h>

#define D 128

typedef __attribute__((ext_vector_type(2))) float v2f;
typedef __attribute__((ext_vector_type(8))) float v8f;

// ---------------------------------------------------------------------------
// GEMM: C[n,128] = A[n,128] @ W[128,128] using V_WMMA_F32_16X16X4_F32.
// One wave per 16-row strip; 8 v8f accumulators cover all 128 output columns.
// ---------------------------------------------------------------------------
__global__ void __launch_bounds__(256) gemm_f32_wmma(
    const float* __restrict__ A, const float* __restrict__ W,
    float* __restrict__ C, int nrows) {
  const int lane  = threadIdx.x & 31;
  const int wave  = threadIdx.x >> 5;
  const int row0  = (blockIdx.x * 8 + wave) * 16;
  if (row0 >= nrows) return;                 // wave-uniform: EXEC stays full
  const int m     = lane & 15;               // M (for A) / N (for B,C)
  const int khalf = lane >> 4;               // 0: K pair {0,1}, 1: K pair {2,3}

  v8f acc[8] = {};

  for (int k = 0; k < D; k += 4) {
    const int kk = k + khalf * 2;
    // A 16x4 f32 layout: lane m holds A[m][kk], A[m][kk+1]
    int ra = row0 + m; if (ra >= nrows) ra = nrows - 1;
    v2f a;
    a.x = A[(long)ra * D + kk];
    a.y = A[(long)ra * D + kk + 1];
#pragma unroll
    for (int j = 0; j < 8; ++j) {
      // B 4x16 f32 layout (mirror of A): lane m holds B[kk][n], B[kk+1][n]
      v2f b;
      b.x = W[(long)(kk + 0) * D + j * 16 + m];
      b.y = W[(long)(kk + 1) * D + j * 16 + m];
      acc[j] = __builtin_amdgcn_wmma_f32_16x16x4_f32(
          false, a, false, b, (short)0, acc[j], false, false);
    }
  }

  // C 16x16 f32 layout: VGPR r, lanes0-15 -> M=r, lanes16-31 -> M=r+8; N=m
#pragma unroll
  for (int j = 0; j < 8; ++j) {
#pragma unroll
    for (int r = 0; r < 8; ++r) {
      const int row = row0 + r + khalf * 8;
      if (row < nrows) C[(long)row * D + j * 16 + m] = acc[j][r];
    }
  }
}

// ---------------------------------------------------------------------------
// Fill helpers (graph-capture-safe scratch init)
// ---------------------------------------------------------------------------
__global__ void fill_f4(float4* __restrict__ p, long n4) {
  long i = (long)blockIdx.x * blockDim.x + threadIdx.x;
  if (i < n4) p[i] = make_float4(0.f, 0.f, 0.f, 0.f);
}

__global__ void fill_u32(unsigned* __restrict__ p, unsigned v, int n) {
  int i = blockIdx.x * blockDim.x + threadIdx.x;
  if (i < n) p[i] = v;
}

// ---------------------------------------------------------------------------
// SpMM scatter: one wave per edge, each lane moves a float4 (128B/wave rows,
// 512B contiguous gather). Native f32 atomics into L2-resident agg buffer.
// ---------------------------------------------------------------------------
__global__ void __launch_bounds__(256) spmm_scatter(
    const float* __restrict__ support, const int* __restrict__ src,
    const int* __restrict__ dst, const float* __restrict__ ew,
    float* __restrict__ agg, int nedges) {
  const int e = (int)(((long)blockIdx.x * blockDim.x + threadIdx.x) >> 5);
  if (e >= nedges) return;
  const int lane = threadIdx.x & 31;
  const int s = src[e];
  const int d = dst[e];
  const float w = ew[e];
  const float4 v = ((const float4*)(support + (long)s * D))[lane];
  float* ap = agg + (long)d * D + lane * 4;
  unsafeAtomicAdd(ap + 0, v.x * w);
  unsafeAtomicAdd(ap + 1, v.y * w);
  unsafeAtomicAdd(ap + 2, v.z * w);
  unsafeAtomicAdd(ap + 3, v.w * w);
}

// ---------------------------------------------------------------------------
// Epilogue: h_out = (relu?) (agg + b); per-column max folded in via a
// monotonic uint encoding + atomicMax (one atomic per thread).
// ---------------------------------------------------------------------------
__device__ __forceinline__ unsigned enc_f32(float f) {
  unsigned b = __float_as_uint(f);
  return (b & 0x80000000u) ? ~b : (b | 0x80000000u);
}
__device__ __forceinline__ float dec_f32(unsigned u) {
  unsigned b = (u & 0x80000000u) ? (u & 0x7FFFFFFFu) : ~u;
  return __uint_as_float(b);
}
#define ENC_NEG_INF 0x007FFFFFu  // enc(-inf)

__global__ void __launch_bounds__(256) bias_act_colmax(
    const float* __restrict__ agg, const float* __restrict__ bias,
    float* __restrict__ hout, unsigned* __restrict__ omax,
    int nrows, int do_relu) {
  const int tid = blockIdx.x * blockDim.x + threadIdx.x;
  const int col = tid & (D - 1);
  const int row0 = tid >> 7;
  const int rstride = (gridDim.x * blockDim.x) >> 7;
  const float b = bias[col];
  float m = -INFINITY;
  for (int r = row0; r < nrows; r += rstride) {
    float v = agg[(long)r * D + col] + b;
    if (do_relu) v = fmaxf(v, 0.f);
    hout[(long)r * D + col] = v;
    m = fmaxf(m, v);
  }
  atomicMax(&omax[col], enc_f32(m));
}

// ---------------------------------------------------------------------------
// Head: lin_in = [o1|o2|o3] (384), logits = lin_W @ lin_in + lin_b, log_softmax
// ---------------------------------------------------------------------------
__global__ void __launch_bounds__(384) head_kernel(
    const unsigned* __restrict__ o1, const unsigned* __restrict__ o2,
    const unsigned* __restrict__ o3, const float* __restrict__ linW,
    const float* __restrict__ linb, float* __restrict__ out) {
  __shared__ float lin_in[3 * D];
  __shared__ float logits[16];
  const int t = threadIdx.x;
  if (t < 3 * D) {
    const unsigned* o = (t < D) ? o1 : ((t < 2 * D) ? o2 : o3);
    lin_in[t] = dec_f32(o[t & (D - 1)]);
  }
  __syncthreads();
  if (t < 10) {
    float acc = linb[t];
    for (int j = 0; j < 3 * D; ++j) acc += linW[t * (3 * D) + j] * lin_in[j];
    logits[t] = acc;
  }
  __syncthreads();
  if (t == 0) {
    float mx = -INFINITY;
    for (int c = 0; c < 10; ++c) mx = fmaxf(mx, logits[c]);
    float s = 0.f;
    for (int c = 0; c < 10; ++c) s += __expf(logits[c] - mx);
    const float lse = __logf(s);
    for (int c = 0; c < 10; ++c) out[c] = logits[c] - mx - lse;
  }
}

// ---------------------------------------------------------------------------
// Launch: 3 layers with rotating N*D workspace buffers, then head.
// ---------------------------------------------------------------------------
static inline void run_layer(const float* h_in, const float* Wm, const float* bv,
                             float* support, float* agg_out, unsigned* omax,
                             int n, int e, int do_relu,
                             const int* esrc, const int* edst, const float* ew,
                             hipStream_t stream) {
  const int tiles = (n + 15) / 16;
  gemm_f32_wmma<<<(tiles + 7) / 8, 256, 0, stream>>>(h_in, Wm, support, n);

  const long n4 = (long)n * D / 4;
  fill_f4<<<(int)((n4 + 255) / 256), 256, 0, stream>>>((float4*)agg_out, n4);

  const long sthreads = (long)e * 32;
  spmm_scatter<<<(int)((sthreads + 255) / 256), 256, 0, stream>>>(
      support, esrc, edst, ew, agg_out, e);

  bias_act_colmax<<<512, 256, 0, stream>>>(agg_out, bv, agg_out, omax, n, do_relu);
}

extern "C" void kernel_launch(void* const* d_in, const int* in_sizes, int n_in,
                              void* d_out, int out_size, void* d_ws, size_t ws_size,
                              hipStream_t stream) {
  const float* x     = (const float*)d_in[0];
  const int*   esrc  = (const int*)d_in[1];
  const int*   edst  = (const int*)d_in[2];
  const float* ew    = (const float*)d_in[3];
  const float* W1    = (const float*)d_in[4];
  const float* b1    = (const float*)d_in[5];
  const float* W2    = (const float*)d_in[6];
  const float* b2    = (const float*)d_in[7];
  const float* W3    = (const float*)d_in[8];
  const float* b3    = (const float*)d_in[9];
  const float* linW  = (const float*)d_in[10];
  const float* linb  = (const float*)d_in[11];

  const int n = in_sizes[0] / D;   // 50000
  const int e = in_sizes[1];       // 800000

  float* buf0 = (float*)d_ws;                       // support scratch
  float* buf1 = buf0 + (size_t)n * D;               // h1 / layer-3 agg
  float* buf2 = buf1 + (size_t)n * D;               // h2
  unsigned* o1 = (unsigned*)(buf2 + (size_t)n * D); // 3*128 encoded maxima
  unsigned* o2 = o1 + D;
  unsigned* o3 = o2 + D;

  fill_u32<<<2, 256, 0, stream>>>(o1, ENC_NEG_INF, 3 * D);

  // layer 1: x -> buf1 (relu), support in buf0
  run_layer(x,    W1, b1, buf0, buf1, o1, n, e, 1, esrc, edst, ew, stream);
  // layer 2: buf1 -> buf2 (relu), support reuses buf0
  run_layer(buf1, W2, b2, buf0, buf2, o2, n, e, 1, esrc, edst, ew, stream);
  // layer 3: buf2 -> buf1 (no relu), support reuses buf0
  run_layer(buf2, W3, b3, buf0, buf1, o3, n, e, 0, esrc, edst, ew, stream);

  head_kernel<<<1, 384, 0, stream>>>(o1, o2, o3, linW, linb, (float*)d_out);
}